// GraphAttConv_88897233093180
// MI455X (gfx1250) — compile-verified
//
#include <hip/hip_runtime.h>
#include <hip/hip_bf16.h>

#define N_IN_F   256
#define N_HEADS  8
#define N_D      32
#define OUT_F    (N_HEADS * N_D)   // 256
#define ALPHA_SLOPE 0.2f
#define EPS_F 1e-16f

typedef __attribute__((ext_vector_type(2))) float v2f;
typedef __attribute__((ext_vector_type(8))) float v8f;

// ---------------------------------------------------------------------------
// monotone float <-> uint encoding for atomicMax-based segment max
// ---------------------------------------------------------------------------
__device__ __forceinline__ unsigned int f2ord(float f) {
    unsigned int u = __float_as_uint(f);
    return (u & 0x80000000u) ? ~u : (u | 0x80000000u);
}
__device__ __forceinline__ float ord2f(unsigned int u) {
    u = (u & 0x80000000u) ? (u & 0x7FFFFFFFu) : ~u;
    return __uint_as_float(u);
}
// f2ord(-inf) = ~bits(-inf) = 0x007FFFFF
#define ORD_NEG_INF 0x007FFFFFu

// ---------------------------------------------------------------------------
// Kernel 0: init out=0, sum=0, mx=-inf(encoded)
// ---------------------------------------------------------------------------
__global__ __launch_bounds__(256)
void gat_init_kernel(float* __restrict__ out, unsigned int* __restrict__ mx,
                     float* __restrict__ ssum, int out_n, int nh) {
    int i = blockIdx.x * blockDim.x + threadIdx.x;
    int stride = gridDim.x * blockDim.x;
    for (int j = i; j < out_n; j += stride) out[j] = 0.0f;
    for (int j = i; j < nh; j += stride) { mx[j] = ORD_NEG_INF; ssum[j] = 0.0f; }
}

// ---------------------------------------------------------------------------
// Kernel 1: h[n, 256] = x[n, 256] @ W_flat[256, 256] via V_WMMA_F32_16X16X4_F32
//   Templated on M-block count so the hot loop is completely branch-free:
//   MB=4 covers all full groups of 4 M-tiles (one B fragment feeds 4 WMMAs),
//   MB=1 covers the remainder tiles in a separate tiny launch.
//   Block = 8 waves -> 8 N-tiles; grid = (2, #groups). No divergence anywhere,
//   so EXEC is all-1s at every WMMA as the ISA requires.
// ---------------------------------------------------------------------------
template <int MB>
__global__ __launch_bounds__(256)
void gat_gemm_wmma_kernel(const float* __restrict__ x, const float* __restrict__ W,
                          float* __restrict__ h, int mtile_base) {
    const int wave = threadIdx.x >> 5;
    const int lane = threadIdx.x & 31;
    const int tileN = blockIdx.x * 8 + wave;     // 0..15
    const int half = lane >> 4;                  // 0: lanes 0-15, 1: lanes 16-31
    const int l15  = lane & 15;
    const int kOff = half << 1;                  // 0 or 2 within the K=4 step
    const int tileM0 = mtile_base + blockIdx.y * MB;

    // B: col = tileN*16 + l15 of W_flat; rows kb+kOff, kb+kOff+1 (stride 32 floats)
    const int nCol = tileN * 16 + l15;
    const int head = nCol >> 5;
    const int dcol = nCol & 31;
    const float* Wcol = W + (size_t)head * N_IN_F * N_D + dcol + (size_t)kOff * N_D;

    // A: rows (tileM0+m)*16 + l15, elements [kb+kOff, kb+kOff+1]
    const float* xrow[MB];
#pragma unroll
    for (int m = 0; m < MB; ++m)
        xrow[m] = x + (size_t)((tileM0 + m) * 16 + l15) * N_IN_F + kOff;

    v8f acc[MB];
#pragma unroll
    for (int m = 0; m < MB; ++m) acc[m] = (v8f){0.f,0.f,0.f,0.f,0.f,0.f,0.f,0.f};

#pragma unroll 2
    for (int kb = 0; kb < N_IN_F; kb += 4) {
        v2f b;
        b.x = Wcol[(size_t)kb * N_D];
        b.y = Wcol[(size_t)(kb + 1) * N_D];
#pragma unroll
        for (int m = 0; m < MB; ++m) {
            v2f a;
            a.x = xrow[m][kb];
            a.y = xrow[m][kb + 1];
            acc[m] = __builtin_amdgcn_wmma_f32_16x16x4_f32(
                /*neg_a=*/false, a, /*neg_b=*/false, b,
                /*c_mod=*/(short)0, acc[m], /*reuse_a=*/false, /*reuse_b=*/false);
        }
    }

    // C/D layout: VGPR j -> row (half*8 + j), col = l15
    const int col = tileN * 16 + l15;
#pragma unroll
    for (int m = 0; m < MB; ++m) {
        const int mbase = (tileM0 + m) * 16 + (half << 3);
#pragma unroll
        for (int j = 0; j < 8; ++j) {
            h[(size_t)(mbase + j) * OUT_F + col] = acc[m][j];
        }
    }
}

// ---------------------------------------------------------------------------
// Kernel 2: per (edge, head): lin -> leakyrelu -> alpha; segment max via atomicMax
//   thread t: edge = t>>3, head = t&7  (8 lanes cover one edge's contiguous 1KB)
// ---------------------------------------------------------------------------
__global__ __launch_bounds__(256)
void gat_edge_alpha_kernel(const float* __restrict__ h, const float* __restrict__ a,
                           const int* __restrict__ src, const int* __restrict__ dst,
                           float* __restrict__ alpha_buf, unsigned int* __restrict__ mx,
                           int n_edges) {
    int t = blockIdx.x * blockDim.x + threadIdx.x;
    int e = t >> 3;
    int head = t & 7;
    if (e >= n_edges) return;

    const int s = src[e];
    const int d = dst[e];
    const float4* hs4 = reinterpret_cast<const float4*>(h + (size_t)s * OUT_F + head * N_D);
    const float4* hd4 = reinterpret_cast<const float4*>(h + (size_t)d * OUT_F + head * N_D);
    const float4* a1 = reinterpret_cast<const float4*>(a + (size_t)head * 5 * N_D + 0 * N_D);
    const float4* a2 = reinterpret_cast<const float4*>(a + (size_t)head * 5 * N_D + 1 * N_D);
    const float4* a3 = reinterpret_cast<const float4*>(a + (size_t)head * 5 * N_D + 2 * N_D);
    const float4* a4 = reinterpret_cast<const float4*>(a + (size_t)head * 5 * N_D + 3 * N_D);
    const float4* a5 = reinterpret_cast<const float4*>(a + (size_t)head * 5 * N_D + 4 * N_D);

    float lin = 0.0f;
#pragma unroll
    for (int i = 0; i < N_D / 4; ++i) {
        float4 vs = hs4[i], vd = hd4[i];
        float4 c1 = a1[i], c2 = a2[i], c3 = a3[i], c4 = a4[i], c5 = a5[i];
        {
            float df = vd.x - vs.x;
            lin += vs.x * c1.x + vd.x * c2.x + df * c3.x + fabsf(df) * c4.x + vs.x * vd.x * c5.x;
        }
        {
            float df = vd.y - vs.y;
            lin += vs.y * c1.y + vd.y * c2.y + df * c3.y + fabsf(df) * c4.y + vs.y * vd.y * c5.y;
        }
        {
            float df = vd.z - vs.z;
            lin += vs.z * c1.z + vd.z * c2.z + df * c3.z + fabsf(df) * c4.z + vs.z * vd.z * c5.z;
        }
        {
            float df = vd.w - vs.w;
            lin += vs.w * c1.w + vd.w * c2.w + df * c3.w + fabsf(df) * c4.w + vs.w * vd.w * c5.w;
        }
    }
    float al = (lin > 0.0f) ? lin : ALPHA_SLOPE * lin;
    alpha_buf[(size_t)e * N_HEADS + head] = al;
    atomicMax(&mx[(size_t)s * N_HEADS + head], f2ord(al));
}

// ---------------------------------------------------------------------------
// Kernel 3: e = exp(alpha - mx[src]); segment sum via atomicAdd; overwrite alpha buf
// ---------------------------------------------------------------------------
__global__ __launch_bounds__(256)
void gat_edge_exp_kernel(const int* __restrict__ src, float* __restrict__ alpha_buf,
                         const unsigned int* __restrict__ mx, float* __restrict__ ssum,
                         int n_edges) {
    int t = blockIdx.x * blockDim.x + threadIdx.x;
    int e = t >> 3;
    int head = t & 7;
    if (e >= n_edges) return;
    const int s = src[e];
    float al = alpha_buf[t];
    float m = ord2f(mx[(size_t)s * N_HEADS + head]);
    float ex = __expf(al - m);
    alpha_buf[t] = ex;
    atomicAdd(&ssum[(size_t)s * N_HEADS + head], ex);
}

// ---------------------------------------------------------------------------
// Kernel 4: out[src] += (e / (sum[src]+eps)) * h[dst]
//   one 256-thread block per edge: thread t -> head t>>5, feature t
//   fully coalesced 1KB load + 1KB of coalesced float atomics
// ---------------------------------------------------------------------------
__global__ __launch_bounds__(256)
void gat_edge_out_kernel(const float* __restrict__ h, const int* __restrict__ src,
                         const int* __restrict__ dst, const float* __restrict__ e_buf,
                         const float* __restrict__ ssum, float* __restrict__ out,
                         int n_edges) {
    int e = blockIdx.x;
    if (e >= n_edges) return;
    int t = threadIdx.x;          // 0..255
    int head = t >> 5;
    const int s = src[e];
    const int d = dst[e];
    float att = e_buf[(size_t)e * N_HEADS + head] /
                (ssum[(size_t)s * N_HEADS + head] + EPS_F);
    float v = h[(size_t)d * OUT_F + t];
    atomicAdd(&out[(size_t)s * OUT_F + t], att * v);
}

// ---------------------------------------------------------------------------
// launch
// ---------------------------------------------------------------------------
extern "C" void kernel_launch(void* const* d_in, const int* in_sizes, int n_in,
                              void* d_out, int out_size, void* d_ws, size_t ws_size,
                              hipStream_t stream) {
    const float* x = (const float*)d_in[0];   // [n, 256]
    const float* W = (const float*)d_in[1];   // [8, 256, 32]
    const float* a = (const float*)d_in[2];   // [8, 160]
    const int* src = (const int*)d_in[3];     // [E]
    const int* dst = (const int*)d_in[4];     // [E]
    float* out = (float*)d_out;               // [n, 256]

    const int n_nodes = in_sizes[0] / N_IN_F; // 50000
    const int n_edges = in_sizes[3];          // 800000

    // workspace carve-up
    float* h = (float*)d_ws;                                     // n*256 floats
    float* alpha_buf = h + (size_t)n_nodes * OUT_F;              // E*8 floats
    unsigned int* mx = (unsigned int*)(alpha_buf + (size_t)n_edges * N_HEADS); // n*8
    float* ssum = (float*)(mx + (size_t)n_nodes * N_HEADS);      // n*8 floats

    // 0: init out / mx / ssum
    {
        int out_n = n_nodes * OUT_F;
        int nh = n_nodes * N_HEADS;
        gat_init_kernel<<<4096, 256, 0, stream>>>(out, mx, ssum, out_n, nh);
    }

    // 1: projection GEMM via WMMA
    //    main: branch-free MB=4 groups; tail: MB=1 for leftover M-tiles
    {
        int mtiles = n_nodes / 16;            // 3125 (n_nodes % 16 == 0)
        int mgroups = mtiles / 4;             // 781 full groups of 4
        int mrem = mtiles - mgroups * 4;      // 1 leftover tile
        if (mgroups > 0) {
            dim3 grid(2, mgroups);
            gat_gemm_wmma_kernel<4><<<grid, 256, 0, stream>>>(x, W, h, 0);
        }
        if (mrem > 0) {
            dim3 grid(2, mrem);
            gat_gemm_wmma_kernel<1><<<grid, 256, 0, stream>>>(x, W, h, mgroups * 4);
        }
    }

    // 2: edge attention logits + segment max
    {
        int total = n_edges * N_HEADS;
        int blocks = (total + 255) / 256;
        gat_edge_alpha_kernel<<<blocks, 256, 0, stream>>>(h, a, src, dst, alpha_buf, mx, n_edges);
    }

    // 3: exp + segment sum
    {
        int total = n_edges * N_HEADS;
        int blocks = (total + 255) / 256;
        gat_edge_exp_kernel<<<blocks, 256, 0, stream>>>(src, alpha_buf, mx, ssum, n_edges);
    }

    // 4: weighted scatter into out
    {
        gat_edge_out_kernel<<<n_edges, 256, 0, stream>>>(h, src, dst, alpha_buf, ssum, out, n_edges);
    }
}